// SoftDTW_Loss_34196529611298
// MI455X (gfx1250) — compile-verified
//
#include <hip/hip_runtime.h>

// ---------------- problem constants (fixed by the reference) ----------------
#define MDIM 4096                      // rows of distance matrix
#define NDIM 4096                      // cols of distance matrix
#define NTHREADS 1024                  // 32 wave32s, one WGP, persistent
#define ROWS_PER_THREAD (MDIM / NTHREADS)   // 4
#define TC 4                           // columns staged per tile
#define RS 6                           // padded floats per row in LDS tile (4 data + 2 pad)
#define NTILES (NDIM / TC)             // 1024
#define LOADS_PER_THREAD (ROWS_PER_THREAD * 2)  // 2x B64 per row -> 8 async ops/thread/tile
#define GAMMA_F 0.01f
#define INV_GAMMA_F 100.0f
#define BIGF 1e30f

#define AS1 __attribute__((address_space(1)))
#define AS3 __attribute__((address_space(3)))

typedef int v2i __attribute__((vector_size(8)));   // matches builtin's pointee type

// ---------------- CDNA5 async global->LDS copy (ASYNCcnt path) ----------------
#if __has_builtin(__builtin_amdgcn_global_load_async_to_lds_b64)
#define ASYNC_B64(g, l, off) do {                                             \
  AS1 void* _g_ = (AS1 void*)(g);                                             \
  AS3 void* _l_ = (AS3 void*)(l);                                             \
  __builtin_amdgcn_global_load_async_to_lds_b64((AS1 v2i*)_g_, (AS3 v2i*)_l_, \
                                                (off), 0);                    \
} while (0)
#else
#define ASYNC_B64(g, l, off) do {                                             \
  unsigned _lo_ = (unsigned)(__SIZE_TYPE__)(AS3 void*)(l);                    \
  asm volatile("global_load_async_to_lds_b64 %0, %1, off offset:%2"           \
               :: "v"(_lo_), "v"((const void*)(g)), "i"(off) : "memory");     \
} while (0)
#endif

#if __has_builtin(__builtin_amdgcn_s_wait_asynccnt)
#define WAIT_ASYNC(n) __builtin_amdgcn_s_wait_asynccnt(n)
#else
#define WAIT_ASYNC(n) asm volatile("s_wait_asynccnt %0" :: "i"(n) : "memory")
#endif

// ---------------- kernel ----------------
// Single persistent workgroup scans columns left->right.
// Recurrence (reference): D[i,j] = C[i-1,j-1] + softmin_g(D[i-1,j-1], D[i,j-1])
// -> each column depends only on the previous column; column itself is parallel.
// dbuf[2][]  : ping-pong DP state D[0..M] for current / next column.
// tile[2][]  : double-buffered staged column tiles (4096 rows x TC cols),
//              row stride RS floats (24B): B64-aligned, LDS bank-conflict-free
//              (6*d ≡ 0 mod 64  =>  d ≡ 0 mod 32: no two of 32 lanes collide).
__global__ __launch_bounds__(NTHREADS, 1)
void softdtw_scan_kernel(const float* __restrict__ dist, float* __restrict__ out) {
  __shared__ float dbuf[2][MDIM + 4];
  __shared__ float tile[2][MDIM * RS];

  const int tid = threadIdx.x;

  // Column 0 state: D[0,0] = 0, D[i,0] = BIG.
  for (int i = tid; i <= MDIM; i += NTHREADS)
    dbuf[0][i] = (i == 0) ? 0.0f : BIGF;

  // Issue async loads for one tile: each thread moves 4 rows x 16B (2x B64).
  auto stage_tile = [&](int t, int buf) {
    const int c0 = t * TC;
    #pragma unroll
    for (int k = 0; k < ROWS_PER_THREAD; ++k) {
      const int r = tid + k * NTHREADS;
      const float* g = dist + (size_t)r * NDIM + c0;   // 16B-aligned (c0 % 4 == 0)
      float* l = &tile[buf][r * RS];                   // 24B row stride, 8B-aligned
      ASYNC_B64(g, l, 0);                              // floats c0..c0+1
      ASYNC_B64(g, l, 8);                              // floats c0+2..c0+3
    }
  };

  stage_tile(0, 0);

  int cur = 0;
  for (int t = 0; t < NTILES; ++t) {
    // Prefetch next tile into the other buffer, then wait for THIS tile's
    // 8 loads (loads complete in order; <= 8 outstanding => tile t done).
    if (t + 1 < NTILES) {
      stage_tile(t + 1, (t + 1) & 1);
      WAIT_ASYNC(LOADS_PER_THREAD);
    } else {
      WAIT_ASYNC(0);
    }
    __syncthreads();   // all waves' async writes for tile t are visible

    const float* tb = tile[t & 1];
    #pragma unroll 1
    for (int jj = 0; jj < TC; ++jj) {
      const float* dc = dbuf[cur];
      float*       dn = dbuf[cur ^ 1];
      #pragma unroll
      for (int k = 0; k < ROWS_PER_THREAD; ++k) {
        const int i = 1 + tid + k * NTHREADS;          // row 1..4096
        const float a  = dc[i - 1];                    // D[i-1, j-1]
        const float b  = dc[i];                        // D[i,   j-1]
        const float mn = fminf(a, b);
        const float df = fabsf(a - b);
        // softmin_g(a,b) = min - g*log(1 + exp(-|a-b|/g))  (== reference's
        // rmax/rsum form; one exp term is always 1, so the 1e-10 clamp is dead)
        const float sm = mn - GAMMA_F * __logf(1.0f + __expf(-df * INV_GAMMA_F));
        dn[i] = tb[(i - 1) * RS + jj] + sm;            // + C[i-1, j-1]
      }
      if (tid == 0) dn[0] = BIGF;                      // D[0, j>=1] = BIG
      __syncthreads();
      cur ^= 1;
    }
  }

  if (tid == 0) out[0] = dbuf[cur][MDIM];              // D[M, N]
}

// ---------------- launch ----------------
extern "C" void kernel_launch(void* const* d_in, const int* in_sizes, int n_in,
                              void* d_out, int out_size, void* d_ws, size_t ws_size,
                              hipStream_t stream) {
  (void)in_sizes; (void)n_in; (void)out_size; (void)d_ws; (void)ws_size;
  const float* dist = (const float*)d_in[0];
  float* out = (float*)d_out;
  softdtw_scan_kernel<<<dim3(1), dim3(NTHREADS), 0, stream>>>(dist, out);
}